// DynamicMemoryWeightsLossAttention_26671746908789
// MI455X (gfx1250) — compile-verified
//
#include <hip/hip_runtime.h>
#include <math.h>

#define NUM_K 64
#define POOL_PER 1024
#define K_SEL 128
#define FEAT 256
#define N_TOK 32768
#define N_COL (NUM_K * POOL_PER) /* 65536 */

typedef float v2f __attribute__((ext_vector_type(2)));
typedef float v8f __attribute__((ext_vector_type(8)));

// ---- helpers -------------------------------------------------------------

// Order-preserving float->uint key (for atomic segment-max over mixed signs).
__device__ __forceinline__ unsigned f2key(float f) {
    unsigned b = __float_as_uint(f);
    return (b & 0x80000000u) ? ~b : (b | 0x80000000u);
}
__device__ __forceinline__ float key2f(unsigned k) {
    unsigned b = (k & 0x80000000u) ? (k ^ 0x80000000u) : ~k;
    return __uint_as_float(b);
}
__device__ __forceinline__ float wave_max(float v) {
    for (int m = 16; m >= 1; m >>= 1) v = fmaxf(v, __shfl_xor(v, m, 32));
    return v;
}
__device__ __forceinline__ float wave_sum(float v) {
    for (int m = 16; m >= 1; m >>= 1) v += __shfl_xor(v, m, 32);
    return v;
}

// ---- kernels -------------------------------------------------------------

__global__ void k_init(unsigned* segkey, float* denom, int* cnt, int* colmap) {
    int i = blockIdx.x * 256 + threadIdx.x;
    if (i < N_COL) colmap[i] = -1;
    if (i < NUM_K * FEAT) { segkey[i] = 0u; denom[i] = 0.0f; }
    if (i < NUM_K) cnt[i] = 0;
}

// Per (f,k) row of 1024 pool values: rowmax and 1/sum(exp(x-max)).
// One wave per row, 32 elements per lane.
__global__ void k_rowstats(const float* __restrict__ pool,
                           float* __restrict__ rowmax, float* __restrict__ rowinv) {
    int wave = (int)((blockIdx.x * blockDim.x + threadIdx.x) >> 5);
    int lane = threadIdx.x & 31;
    if (wave >= FEAT * NUM_K) return;
    int f = wave >> 6, k = wave & 63;
    const float* row = pool + (size_t)f * N_COL + (size_t)k * POOL_PER;
    float x[32];
    float mx = -INFINITY;
#pragma unroll
    for (int i = 0; i < 32; ++i) { x[i] = row[lane + 32 * i]; mx = fmaxf(mx, x[i]); }
    mx = wave_max(mx);
    float s = 0.0f;
#pragma unroll
    for (int i = 0; i < 32; ++i) s += expf(x[i] - mx);
    s = wave_sum(s);
    if (lane == 0) { rowmax[wave] = mx; rowinv[wave] = 1.0f / s; }
}

// s1[c] = sum_f softmax(pool)[f,c], computed with V_WMMA_F32_16X16X4_F32:
// D = A(16 cols x 4 f) * ones(4x16) + C accumulates column sums over f.
// One wave handles 16 consecutive pool columns (all inside one k-band).
__global__ void k_s1(const float* __restrict__ pool,
                     const float* __restrict__ rowmax, const float* __restrict__ rowinv,
                     float* __restrict__ s1) {
    int wave = (int)((blockIdx.x * blockDim.x + threadIdx.x) >> 5);
    int lane = threadIdx.x & 31;
    if (wave >= N_COL / 16) return;
    int col0 = wave * 16;
    int k = col0 >> 10;
    int m = lane & 15;          // M index (column within 16-block)
    int q = (lane >> 4) * 2;    // K slice per ISA A-layout: lanes0-15 -> K={0,1}, 16-31 -> K={2,3}
    v8f c = {};
    v2f b = {1.0f, 1.0f};       // all-ones B => D[m,n] = sum_K A[m,K]
    for (int f = 0; f < FEAT; f += 4) {
        int r0 = (f + q) * NUM_K + k;
        int r1 = r0 + NUM_K;
        float p0 = pool[(size_t)(f + q) * N_COL + col0 + m];
        float p1 = pool[(size_t)(f + q + 1) * N_COL + col0 + m];
        v2f a;
        a.x = expf(p0 - rowmax[r0]) * rowinv[r0];
        a.y = expf(p1 - rowmax[r1]) * rowinv[r1];
        c = __builtin_amdgcn_wmma_f32_16x16x4_f32(false, a, false, b,
                                                  (short)0, c, false, false);
    }
    // D[m, n] identical for all n. Lane 0 holds D[0..7,0], lane 16 holds D[8..15,0].
    if (lane == 0) {
        float4* o = (float4*)(s1 + col0);
        o[0] = make_float4(c[0], c[1], c[2], c[3]);
        o[1] = make_float4(c[4], c[5], c[6], c[7]);
    } else if (lane == 16) {
        float4* o = (float4*)(s1 + col0 + 8);
        o[0] = make_float4(c[0], c[1], c[2], c[3]);
        o[1] = make_float4(c[4], c[5], c[6], c[7]);
    }
}

// Per cluster: bitonic sort 1024 s1 values ascending by (value, index);
// first 128 give idx1 exactly in JAX top_k(-s1) order.
__global__ __launch_bounds__(1024) void k_sort1(const float* __restrict__ s1,
                                                int* __restrict__ colidx) {
    __shared__ float sv[1024];
    __shared__ int   si[1024];
    int k = blockIdx.x, t = threadIdx.x;
    sv[t] = s1[k * 1024 + t];
    si[t] = t;
    __syncthreads();
    for (int sz = 2; sz <= 1024; sz <<= 1) {
        for (int j = sz >> 1; j > 0; j >>= 1) {
            int ixj = t ^ j;
            if (ixj > t) {
                bool up = ((t & sz) == 0);
                float v1 = sv[t], v2 = sv[ixj];
                int i1 = si[t], i2 = si[ixj];
                bool less = (v1 < v2) || (v1 == v2 && i1 < i2);
                if (up ? !less : less) {
                    sv[t] = v2; sv[ixj] = v1; si[t] = i2; si[ixj] = i1;
                }
            }
            __syncthreads();
        }
    }
    if (t < K_SEL) colidx[k * K_SEL + t] = k * POOL_PER + si[t];
}

__global__ void k_segmax(const float* __restrict__ act, const int* __restrict__ clus,
                         unsigned* __restrict__ segkey) {
    int e = blockIdx.x * 256 + threadIdx.x;
    if (e >= N_TOK * FEAT) return;
    int t = e >> 8, j = e & 255;
    int k = clus[t];
    atomicMax(&segkey[k * FEAT + j], f2key(act[e]));
}

__global__ void k_denom(const float* __restrict__ act, const int* __restrict__ clus,
                        const unsigned* __restrict__ segkey, float* __restrict__ denom) {
    int e = blockIdx.x * 256 + threadIdx.x;
    if (e >= N_TOK * FEAT) return;
    int t = e >> 8, j = e & 255;
    int k = clus[t];
    float mx = key2f(segkey[k * FEAT + j]);
    atomicAdd(&denom[k * FEAT + j], expf(act[e] - mx));
}

// score[t] = sum_j exp(a - segmax)/denom. One wave per token.
__global__ void k_score(const float* __restrict__ act, const int* __restrict__ clus,
                        const unsigned* __restrict__ segkey, const float* __restrict__ denom,
                        float* __restrict__ score) {
    int wave = (int)((blockIdx.x * blockDim.x + threadIdx.x) >> 5);
    int lane = threadIdx.x & 31;
    if (wave >= N_TOK) return;
    int t = wave, k = clus[t];
    float s = 0.0f;
#pragma unroll
    for (int i = 0; i < 8; ++i) {
        int j = lane + 32 * i;
        float mx = key2f(segkey[k * FEAT + j]);
        s += expf(act[(size_t)t * FEAT + j] - mx) / denom[k * FEAT + j];
    }
    s = wave_sum(s);
    if (lane == 0) score[t] = s;
}

__global__ void k_scatter(const int* __restrict__ clus, int* __restrict__ cnt,
                          int* __restrict__ list) {
    int t = blockIdx.x * 256 + threadIdx.x;
    if (t >= N_TOK) return;
    int k = clus[t];
    int p = atomicAdd(&cnt[k], 1);
    if (p < POOL_PER) list[k * POOL_PER + p] = t;
}

// Per cluster: sort member tokens by (score desc, index asc); first 128 = idx2.
__global__ __launch_bounds__(1024) void k_sort2(const float* __restrict__ score,
                                                const int* __restrict__ cnt,
                                                const int* __restrict__ list,
                                                int* __restrict__ idx2tok) {
    __shared__ float sv[1024];
    __shared__ int   si[1024];
    int k = blockIdx.x, t = threadIdx.x;
    int c = cnt[k]; if (c > 1024) c = 1024;
    float v; int id;
    if (t < c) { id = list[k * 1024 + t]; v = score[id]; }
    else       { id = 0x7FFFFFFF;         v = -INFINITY; }
    sv[t] = v; si[t] = id;
    __syncthreads();
    for (int sz = 2; sz <= 1024; sz <<= 1) {
        for (int j = sz >> 1; j > 0; j >>= 1) {
            int ixj = t ^ j;
            if (ixj > t) {
                bool up = ((t & sz) == 0);
                float v1 = sv[t], v2 = sv[ixj];
                int i1 = si[t], i2 = si[ixj];
                // "ascending" under (value desc, index asc)
                bool less = (v1 > v2) || (v1 == v2 && i1 < i2);
                if (up ? !less : less) {
                    sv[t] = v2; sv[ixj] = v1; si[t] = i2; si[ixj] = i1;
                }
            }
            __syncthreads();
        }
    }
    if (t < K_SEL) idx2tok[k * K_SEL + t] = si[t];
}

__global__ void k_colmap(const int* __restrict__ colidx, int* __restrict__ colmap) {
    int i = blockIdx.x * 256 + threadIdx.x;
    if (i < NUM_K * K_SEL) colmap[colidx[i]] = i;
}

// Fused: out = pool, with selected columns blended with gathered activation rows.
__global__ void k_final(const float* __restrict__ pool, const float* __restrict__ act,
                        const int* __restrict__ colmap, const int* __restrict__ idx2tok,
                        float* __restrict__ out) {
    size_t e = (size_t)blockIdx.x * 256 + threadIdx.x;
    int c = (int)(e & (N_COL - 1));
    int f = (int)(e >> 16);
    float v = pool[e];
    int i = colmap[c];
    if (i >= 0) {
        int tok = idx2tok[i];
        v = 0.7f * v + 0.3f * act[(size_t)tok * FEAT + f];
    }
    out[e] = v;
}

// ---- launch --------------------------------------------------------------

extern "C" void kernel_launch(void* const* d_in, const int* in_sizes, int n_in,
                              void* d_out, int out_size, void* d_ws, size_t ws_size,
                              hipStream_t stream) {
    const float* act  = (const float*)d_in[0];  // (32768, 256)
    const float* pool = (const float*)d_in[1];  // (256, 65536)
    const int*   clus = (const int*)d_in[2];    // (32768,)
    float* out = (float*)d_out;                 // (256, 65536)
    char* ws = (char*)d_ws;

    float*    rowmax  = (float*)(ws + 0);        // 16384 f32
    float*    rowinv  = (float*)(ws + 65536);    // 16384 f32
    float*    s1      = (float*)(ws + 131072);   // 65536 f32
    unsigned* segkey  = (unsigned*)(ws + 393216);// 16384 u32
    float*    denom   = (float*)(ws + 458752);   // 16384 f32
    float*    score   = (float*)(ws + 524288);   // 32768 f32
    int*      cnt     = (int*)(ws + 655360);     // 64
    int*      list    = (int*)(ws + 655616);     // 65536 int
    int*      colidx  = (int*)(ws + 917760);     // 8192 int
    int*      idx2tok = (int*)(ws + 950528);     // 8192 int
    int*      colmap  = (int*)(ws + 983296);     // 65536 int (ends ~1.22 MB)

    k_init    <<<256, 256, 0, stream>>>(segkey, denom, cnt, colmap);
    k_rowstats<<<2048, 256, 0, stream>>>(pool, rowmax, rowinv);
    k_s1      <<<512, 256, 0, stream>>>(pool, rowmax, rowinv, s1);
    k_sort1   <<<64, 1024, 0, stream>>>(s1, colidx);
    k_segmax  <<<32768, 256, 0, stream>>>(act, clus, segkey);
    k_denom   <<<32768, 256, 0, stream>>>(act, clus, segkey, denom);
    k_score   <<<4096, 256, 0, stream>>>(act, clus, segkey, denom, score);
    k_scatter <<<128, 256, 0, stream>>>(clus, cnt, list);
    k_sort2   <<<64, 1024, 0, stream>>>(score, cnt, list, idx2tok);
    k_colmap  <<<32, 256, 0, stream>>>(colidx, colmap);
    k_final   <<<65536, 256, 0, stream>>>(pool, act, colmap, idx2tok, out);
}